// BlazeEarDetectionsWrapper_90022514524899
// MI455X (gfx1250) — compile-verified
//
#include <hip/hip_runtime.h>
#include <hip/hip_bf16.h>

typedef unsigned int       u32;
typedef unsigned long long u64;

typedef __attribute__((ext_vector_type(2))) float v2f;
typedef __attribute__((ext_vector_type(8))) float v8f;

#define N_ELEM      4194304
#define NVEC        (N_ELEM / 4)
#define TOPK        1024
#define IOU_THRESH  0.3f
#define CLIPV       100.0f
#define INV_SCALE   0.0078125f   /* 1/128 exact */
#define NBINS       2048
#define CAND_CAP    32768

/* ---- workspace layout (bytes) ---- */
#define HIST_OFF    0            /* 2048 u32            */
#define CTRL_OFF    8192         /* 16 u32: [0]=T [2]=candCount */
#define MASK_OFF    8448         /* 1024*32 u32 = 128KB */
#define ZERO_BYTES  139520       /* memset range: hist+ctrl+mask */
#define CAND_OFF    139520       /* 32768 u64 = 256KB   */
#define TKEY_OFF    401664       /* 1024 u64            */
#define BOX_OFF     409856       /* 1024*4 f32 (yxyx)   */
#define SCORE_OFF   426240       /* 1024 f32            */
#define AREA_OFF    430336       /* 1024 f32            */
#define KEEP_OFF    434432       /* 1024 u32            */

/* ---- CDNA5 feature probes ---- */
#if __has_builtin(__builtin_amdgcn_global_load_async_to_lds_b128)
#define HAVE_ASYNC_LDS 1
/* builtin signature (from clang diagnostic): (v4i AS1*, v4i AS3*, imm, imm) */
typedef int v4i_t __attribute__((vector_size(16)));
typedef __attribute__((address_space(1))) v4i_t* g_v4p;
typedef __attribute__((address_space(3))) v4i_t* l_v4p;
#endif

#if __has_builtin(__builtin_amdgcn_s_wait_asynccnt)
#define WAIT_ASYNC(n) __builtin_amdgcn_s_wait_asynccnt(n)
#else
#define WAIT_ASYNC(n) asm volatile("s_wait_asynccnt %0" :: "n"(n) : "memory")
#endif

#if __has_builtin(__builtin_amdgcn_wmma_f32_16x16x4_f32)
#define HAVE_WMMA_F32X4 1
#endif

__device__ __forceinline__ u32 score_key(float x) {
  u32 u = __float_as_uint(x);
  /* negatives -> 0 (score thresholded to 0); x>=0 -> bits+1 (monotone) */
  return (u & 0x80000000u) ? 0u : (u + 1u);
}

/* ================= Pass 1: 2048-bin histogram of key[31:21] ============== */
#define HIST_GRID   1024
#define HIST_BLK    256

__global__ void hist_kernel(const float* __restrict__ scores, u32* __restrict__ hist) {
  __shared__ u32 lhist[NBINS];
#if HAVE_ASYNC_LDS
  __shared__ __align__(16) float4 stage[2][HIST_BLK];
#endif
  for (int b = threadIdx.x; b < NBINS; b += blockDim.x) lhist[b] = 0u;
  __syncthreads();

  const int stride = HIST_GRID * HIST_BLK;                 /* 262144 vec4 */
  const int base   = blockIdx.x * HIST_BLK + threadIdx.x;
  const int nstage = NVEC / stride;                        /* 4 exactly   */
  const float4* gv = (const float4*)scores;

#if HAVE_ASYNC_LDS
  /* double-buffered async global->LDS streaming, tracked by ASYNCcnt */
  __builtin_amdgcn_global_load_async_to_lds_b128(
      (g_v4p)(gv + base), (l_v4p)&stage[0][threadIdx.x], 0, 0);
  for (int s = 0; s < nstage; ++s) {
    if (s + 1 < nstage) {
      __builtin_amdgcn_global_load_async_to_lds_b128(
          (g_v4p)(gv + base + (s + 1) * stride),
          (l_v4p)&stage[(s + 1) & 1][threadIdx.x], 0, 0);
      WAIT_ASYNC(1);
    } else {
      WAIT_ASYNC(0);
    }
    float4 v = stage[s & 1][threadIdx.x];
    atomicAdd(&lhist[score_key(v.x) >> 21], 1u);
    atomicAdd(&lhist[score_key(v.y) >> 21], 1u);
    atomicAdd(&lhist[score_key(v.z) >> 21], 1u);
    atomicAdd(&lhist[score_key(v.w) >> 21], 1u);
  }
#else
  for (int s = 0; s < nstage; ++s) {
    float4 v = gv[base + s * stride];
    atomicAdd(&lhist[score_key(v.x) >> 21], 1u);
    atomicAdd(&lhist[score_key(v.y) >> 21], 1u);
    atomicAdd(&lhist[score_key(v.z) >> 21], 1u);
    atomicAdd(&lhist[score_key(v.w) >> 21], 1u);
  }
#endif
  __syncthreads();
  for (int b = threadIdx.x; b < NBINS; b += blockDim.x) {
    u32 c = lhist[b];
    if (c) atomicAdd(&hist[b], c);
  }
}

/* ================= Pass 2: find threshold bin T ========================= */
__global__ void select_kernel(const u32* __restrict__ hist, u32* __restrict__ ctrl) {
  if (threadIdx.x == 0) {
    u32 cum = 0; u32 T = 0;
    for (int b = NBINS - 1; b >= 0; --b) {
      cum += hist[b];
      if (cum >= (u32)TOPK) { T = (u32)b; break; }
    }
    ctrl[0] = T;
  }
}

/* ================= Pass 3: compact candidates (bin >= T) ================ */
__global__ void compact_kernel(const float* __restrict__ scores,
                               u32* __restrict__ ctrl, u64* __restrict__ cand) {
  const u32 T = ctrl[0];
  const int stride = gridDim.x * blockDim.x;
  const float4* gv = (const float4*)scores;
  for (int v = blockIdx.x * blockDim.x + threadIdx.x; v < NVEC; v += stride) {
    __builtin_prefetch(gv + v + stride, 0, 0);   /* global_prefetch_b8 */
    float4 x = gv[v];
    float  f[4] = {x.x, x.y, x.z, x.w};
#pragma unroll
    for (int c = 0; c < 4; ++c) {
      u32 key = score_key(f[c]);
      if ((key >> 21) >= T) {
        u32 pos = atomicAdd(&ctrl[2], 1u);
        if (pos < CAND_CAP) {
          u32 idx = (u32)(4 * v + c);
          cand[pos] = ((u64)key << 22) | (u64)(0x3FFFFFu - idx);
        }
      }
    }
  }
}

/* ================= Pass 4: exact rank selection of top-1024 ============= */
__global__ void rank_kernel(const u64* __restrict__ cand,
                            const u32* __restrict__ ctrl, u64* __restrict__ tkey) {
  __shared__ u32 sM;
  if (threadIdx.x == 0) {
    u32 m = ctrl[2];
    sM = (m < (u32)CAND_CAP) ? m : (u32)CAND_CAP;
  }
  for (int r = threadIdx.x; r < TOPK; r += blockDim.x) tkey[r] = 0ull;
  __syncthreads();
  const u32 M = sM;
  for (u32 c = threadIdx.x; c < M; c += blockDim.x) {
    u64 k = cand[c];
    u32 rank = 0;
    for (u32 j = 0; j < M; ++j) rank += (cand[j] > k) ? 1u : 0u;   /* keys distinct */
    if (rank < (u32)TOPK) tkey[rank] = k;
  }
}

/* ================= Pass 5: gather + decode 1024 boxes =================== */
__global__ void decode_kernel(const float* __restrict__ raw_boxes,
                              const float* __restrict__ raw_scores,
                              const float* __restrict__ anchors,
                              const u64* __restrict__ tkey,
                              float* __restrict__ box, float* __restrict__ score,
                              float* __restrict__ area) {
  int r = blockIdx.x * blockDim.x + threadIdx.x;
  if (r >= TOPK) return;
  u64 k    = tkey[r];
  u32 k32  = (u32)(k >> 22);
  u32 idx  = 0x3FFFFFu - (u32)(k & 0x3FFFFFu);
  float4 b = ((const float4*)raw_boxes)[idx];
  float4 a = ((const float4*)anchors)[idx];
  float xc = b.x * INV_SCALE * a.z + a.x;
  float yc = b.y * INV_SCALE * a.w + a.y;
  float w  = b.z * INV_SCALE * a.z;
  float h  = b.w * INV_SCALE * a.w;
  float y0 = yc - h * 0.5f, y1 = yc + h * 0.5f;
  float x0 = xc - w * 0.5f, x1 = xc + w * 0.5f;
  float ymin = fminf(y0, y1), ymax = fmaxf(y0, y1);
  float xmin = fminf(x0, x1), xmax = fmaxf(x0, x1);
  box[4 * r + 0] = ymin; box[4 * r + 1] = xmin;
  box[4 * r + 2] = ymax; box[4 * r + 3] = xmax;
  float s = 0.0f;
  if (k32 != 0u) {
    float x = fminf(fmaxf(raw_scores[idx], -CLIPV), CLIPV);
    s = 1.0f / (1.0f + expf(-x));
    if (s < 0.5f) s = 0.0f;
  }
  score[r] = s;
  area[r]  = (xmax - xmin) * (ymax - ymin);
}

/* ====== Pass 6: suppression bit-matrix; WMMA computes area_i+area_j ===== */
/* One wave per 16x16 (i,j) tile: 64x64 tiles = 4096 waves.                 */
__global__ void mask_kernel(const float* __restrict__ box,
                            const float* __restrict__ area,
                            u32* __restrict__ mask) {
  int wave = (blockIdx.x * blockDim.x + threadIdx.x) >> 5;   /* 0..4095 */
  int lane = threadIdx.x & 31;
  int ti = wave >> 6, tj = wave & 63;
  int hi  = lane >> 4;        /* 0: lanes 0-15, 1: lanes 16-31 */
  int l15 = lane & 15;

  float areaI = area[ti * 16 + l15];
  float areaJ = area[tj * 16 + l15];

#if HAVE_WMMA_F32X4
  /* A (16x4): row M=lane, lanes0-15 hold K=0,1 ; lanes16-31 hold K=2,3 (zeros)
     A row = [area_i, 1, 0, 0];  B col = [1, area_j, 0, 0]^T
     => C[M][N] = area_i(M) + area_j(N), exact f32 adds                     */
  v2f av, bv;
  av.x = hi ? 0.0f : areaI;  av.y = hi ? 0.0f : 1.0f;
  bv.x = hi ? 0.0f : 1.0f;   bv.y = hi ? 0.0f : areaJ;
  v8f c = {};
  c = __builtin_amdgcn_wmma_f32_16x16x4_f32(false, av, false, bv,
                                            (short)0, c, false, false);
#endif

  int j = tj * 16 + l15;
  float4 bj = ((const float4*)box)[j];        /* (ymin,xmin,ymax,xmax) */
  float x1j = bj.y, y1j = bj.x, x2j = bj.w, y2j = bj.z;   /* xyxy view */

#pragma unroll
  for (int v = 0; v < 8; ++v) {
    int ia = ti * 16 + v;          /* row for lanes 0-15  */
    int ib = ia + 8;               /* row for lanes 16-31 */
    int i  = hi ? ib : ia;
    float4 bi = ((const float4*)box)[i];
    float x1i = bi.y, y1i = bi.x, x2i = bi.w, y2i = bi.z;
    float ix1 = fmaxf(x1i, x1j), iy1 = fmaxf(y1i, y1j);
    float ix2 = fminf(x2i, x2j), iy2 = fminf(y2i, y2j);
    float inter = fmaxf(ix2 - ix1, 0.0f) * fmaxf(iy2 - iy1, 0.0f);
#if HAVE_WMMA_F32X4
    float asum = c[v];             /* lane layout: M = v + 8*hi, N = l15 */
#else
    float asum = area[i] + areaJ;
#endif
    float uni = asum - inter;
    float iou = (uni > 0.0f) ? (inter / uni) : 0.0f;
    bool pred = (iou > IOU_THRESH) && (j > i);
    u32 bits = (u32)__ballot(pred);            /* wave32: 32-bit mask */
    if (lane == 0) {
      int word = tj >> 1, sh = (tj & 1) << 4;
      atomicOr(&mask[ia * 32 + word], (bits & 0xFFFFu) << sh);
      atomicOr(&mask[ib * 32 + word], (bits >> 16) << sh);
    }
  }
}

/* ====== Pass 7: serial greedy scan over 1024-bit masks (one wave) ======= */
__global__ void scan_kernel(const u32* __restrict__ mask, u32* __restrict__ keep) {
  int w = threadIdx.x;            /* word 0..31 */
  u32 rem = 0;
  for (int i = 0; i < TOPK; ++i) {
    u32 ow = (u32)__shfl((int)rem, i >> 5, 32);
    bool alive = ((ow >> (i & 31)) & 1u) == 0u;
    if (w == 0) keep[i] = alive ? 1u : 0u;
    if (alive) rem |= mask[i * 32 + w];
  }
}

/* ================= Pass 8: write (yxyx, score) * keep =================== */
__global__ void final_kernel(const float* __restrict__ box,
                             const float* __restrict__ score,
                             const u32* __restrict__ keep,
                             float* __restrict__ out) {
  int r = blockIdx.x * blockDim.x + threadIdx.x;
  if (r >= TOPK) return;
  float m = keep[r] ? 1.0f : 0.0f;
  out[5 * r + 0] = box[4 * r + 0] * m;
  out[5 * r + 1] = box[4 * r + 1] * m;
  out[5 * r + 2] = box[4 * r + 2] * m;
  out[5 * r + 3] = box[4 * r + 3] * m;
  out[5 * r + 4] = score[r] * m;
}

extern "C" void kernel_launch(void* const* d_in, const int* in_sizes, int n_in,
                              void* d_out, int out_size, void* d_ws, size_t ws_size,
                              hipStream_t stream) {
  const float* raw_boxes  = (const float*)d_in[0];
  const float* raw_scores = (const float*)d_in[1];
  const float* anchors    = (const float*)d_in[2];
  float* out = (float*)d_out;
  char*  ws  = (char*)d_ws;

  u32* hist  = (u32*)(ws + HIST_OFF);
  u32* ctrl  = (u32*)(ws + CTRL_OFF);
  u32* maskm = (u32*)(ws + MASK_OFF);
  u64* cand  = (u64*)(ws + CAND_OFF);
  u64* tkey  = (u64*)(ws + TKEY_OFF);
  float* box   = (float*)(ws + BOX_OFF);
  float* score = (float*)(ws + SCORE_OFF);
  float* area  = (float*)(ws + AREA_OFF);
  u32* keep  = (u32*)(ws + KEEP_OFF);

  (void)in_sizes; (void)n_in; (void)out_size; (void)ws_size;

  (void)hipMemsetAsync(d_ws, 0, ZERO_BYTES, stream);           /* hist+ctrl+mask */
  hist_kernel   <<<HIST_GRID, HIST_BLK, 0, stream>>>(raw_scores, hist);
  select_kernel <<<1, 32, 0, stream>>>(hist, ctrl);
  compact_kernel<<<2048, 256, 0, stream>>>(raw_scores, ctrl, cand);
  rank_kernel   <<<1, 1024, 0, stream>>>(cand, ctrl, tkey);
  decode_kernel <<<(TOPK + 255) / 256, 256, 0, stream>>>(raw_boxes, raw_scores,
                                                         anchors, tkey, box, score, area);
  mask_kernel   <<<512, 256, 0, stream>>>(box, area, maskm);   /* 4096 waves */
  scan_kernel   <<<1, 32, 0, stream>>>(maskm, keep);
  final_kernel  <<<(TOPK + 255) / 256, 256, 0, stream>>>(box, score, keep, out);
}